// IndexPut1DIntAccumulateModule_39444979647275
// MI455X (gfx1250) — compile-verified
//
#include <hip/hip_runtime.h>

typedef int   int4v   __attribute__((ext_vector_type(4)));
typedef float float4v __attribute__((ext_vector_type(4)));

// ---- Hot scatter kernel (listed first so the disasm snippet shows it). ----
// Straight-line, no bounds check: host launches exactly covered elements.
// 2x global_load_b128 nt  (stream-once index/value, keep out of L2)
// 4x global_atomic_add_f32, result discarded -> non-returning form.
// Unsigned element indices -> SADDR(base)+VGPR32 offset addressing.
__global__ __launch_bounds__(256) void
scatter_add_vec4(const int* __restrict__ idx, const int* __restrict__ val,
                 float* __restrict__ out)
{
    unsigned i = (blockIdx.x * 256u + threadIdx.x) * 4u;
    int4v ix = __builtin_nontemporal_load((const int4v*)(idx + i));
    int4v v  = __builtin_nontemporal_load((const int4v*)(val + i));
    __hip_atomic_fetch_add(&out[(unsigned)ix.x], (float)v.x, __ATOMIC_RELAXED, __HIP_MEMORY_SCOPE_AGENT);
    __hip_atomic_fetch_add(&out[(unsigned)ix.y], (float)v.y, __ATOMIC_RELAXED, __HIP_MEMORY_SCOPE_AGENT);
    __hip_atomic_fetch_add(&out[(unsigned)ix.z], (float)v.z, __ATOMIC_RELAXED, __HIP_MEMORY_SCOPE_AGENT);
    __hip_atomic_fetch_add(&out[(unsigned)ix.w], (float)v.w, __ATOMIC_RELAXED, __HIP_MEMORY_SCOPE_AGENT);
}

// Scalar tail for the last (m % 1024) updates; launched only if needed.
__global__ __launch_bounds__(256) void
scatter_add_tail(const int* __restrict__ idx, const int* __restrict__ val,
                 float* __restrict__ out, int start, int m)
{
    int k = start + (int)(blockIdx.x * 256u + threadIdx.x);
    if (k < m) {
        __hip_atomic_fetch_add(&out[(unsigned)idx[k]], (float)val[k],
                               __ATOMIC_RELAXED, __HIP_MEMORY_SCOPE_AGENT);
    }
}

// ---- Init: out[i] = (float)input[i] ----
// NT load of the stream-once input; regular store so the 64MB atomic target
// lands hot in the 192MB L2 before the scatter pass.
__global__ __launch_bounds__(256) void
init_out_vec4(const int* __restrict__ in, float* __restrict__ out)
{
    unsigned i = (blockIdx.x * 256u + threadIdx.x) * 4u;
    int4v v = __builtin_nontemporal_load((const int4v*)(in + i));
    float4v f;
    f.x = (float)v.x;
    f.y = (float)v.y;
    f.z = (float)v.z;
    f.w = (float)v.w;
    *(float4v*)(out + i) = f;
}

__global__ __launch_bounds__(256) void
init_out_tail(const int* __restrict__ in, float* __restrict__ out, int start, int n)
{
    int k = start + (int)(blockIdx.x * 256u + threadIdx.x);
    if (k < n)
        out[k] = (float)in[k];
}

extern "C" void kernel_launch(void* const* d_in, const int* in_sizes, int n_in,
                              void* d_out, int out_size, void* d_ws, size_t ws_size,
                              hipStream_t stream)
{
    const int* input = (const int*)d_in[0];   // N int32
    const int* index = (const int*)d_in[1];   // M int32
    const int* value = (const int*)d_in[2];   // M int32
    float*     out   = (float*)d_out;         // N float32

    const int n = in_sizes[0];
    const int m = in_sizes[1];

    const int threads = 256;
    const int elemsPerBlock = threads * 4;    // 1024

    // Pass 1: initialize output table.
    int nVec = n & ~(elemsPerBlock - 1);
    if (nVec > 0)
        init_out_vec4<<<nVec / elemsPerBlock, threads, 0, stream>>>(input, out);
    if (n > nVec)
        init_out_tail<<<(n - nVec + threads - 1) / threads, threads, 0, stream>>>(
            input, out, nVec, n);

    // Pass 2: scatter-accumulate.
    int mVec = m & ~(elemsPerBlock - 1);
    if (mVec > 0)
        scatter_add_vec4<<<mVec / elemsPerBlock, threads, 0, stream>>>(index, value, out);
    if (m > mVec)
        scatter_add_tail<<<(m - mVec + threads - 1) / threads, threads, 0, stream>>>(
            index, value, out, mVec, m);
}